// GCN_H_10230612099510
// MI455X (gfx1250) — compile-verified
//
#include <hip/hip_runtime.h>
#include <math.h>
#include <cstdint>

// ---------------------------------------------------------------------------
// GCN + SAGPooling pipeline for MI455X (gfx1250, wave32).
// GEMMs: V_WMMA_F32_16X16X4_F32 fed from LDS, double-buffered with
// GLOBAL_LOAD_ASYNC_TO_LDS_B128 (ASYNCcnt) per CDNA5 ISA ch.10 / 7.12.
// ---------------------------------------------------------------------------

typedef __attribute__((ext_vector_type(2))) float v2f;
typedef __attribute__((ext_vector_type(8))) float v8f;

#define NGCN 4096
#define FDIM 500
#define FDIMP 512   // K padded to a multiple of BK
#define HDIM 256

// GEMM blocking
#define BM 128
#define BN 64
#define BK 32
#define BKP 36                        // padded LDS stride (bank-conflict free,
                                      // keeps 16B alignment: 36*4 = 144)
#define AS_STRIDE (BM * BKP)          // 4608 floats per A buffer
#define BS_OFF (2 * AS_STRIDE)        // 9216
#define BS_STRIDE 2304                // max(64*36, 32*64)
#define SMEM_FLOATS (BS_OFF + 2 * BS_STRIDE)  // 13824 floats = 54 KiB

// ---- CDNA5 async global->LDS copy (16B per lane), ASYNCcnt-tracked --------
__device__ __forceinline__ void async_cp16(uint32_t lds_off, const float* g) {
  // VDST = per-lane LDS byte offset, VADDR = 64-bit global address.
  asm volatile("global_load_async_to_lds_b128 %0, %1, off"
               :: "v"(lds_off), "v"(g) : "memory");
}
__device__ __forceinline__ void async_wait0() {
#if __has_builtin(__builtin_amdgcn_s_wait_asynccnt)
  __builtin_amdgcn_s_wait_asynccnt(0);
#else
  asm volatile("s_wait_asynccnt 0x0" ::: "memory");
#endif
}
__device__ __forceinline__ uint32_t lds_off_u32(const void* p) {
  // generic LDS address: low 32 bits are the wave-relative LDS byte offset
  return (uint32_t)(uintptr_t)p;
}

// ---------------------------------------------------------------------------
// C[M,N] = A[M,K] @ B[K,N] (row-major), fp32 WMMA, LDS double buffered.
//   BT=1  : B accessed transposed (B[n,k])  -> X @ X^T
//   EPI=1 : GCN epilogue C = relu(dvec[m]*(acc + B[m,n]) + bias[n])
// Requires M%128==0, N%64==0, K%32==0. Block = 256 thr (8 waves), each wave
// computes a 32x32 sub-tile (2x2 WMMA accumulators).
// ---------------------------------------------------------------------------
template <bool BT, int EPI>
__global__ __launch_bounds__(256)
void wmma_gemm_f32(const float* __restrict__ A, int lda,
                   const float* __restrict__ B, int ldb,
                   float* __restrict__ C, int ldc,
                   int K,
                   const float* __restrict__ dvec,
                   const float* __restrict__ bias) {
  __shared__ float smem[SMEM_FLOATS];
  const int tid = threadIdx.x;
  const int lane = tid & 31;
  const int wv = tid >> 5;     // 0..7
  const int wr = wv >> 1;      // 0..3 : 32-row strip
  const int wc = wv & 1;       // 0..1 : 32-col strip
  const int lr = lane & 15;
  const int lh = lane >> 4;
  const int bm0 = blockIdx.y * BM;
  const int bn0 = blockIdx.x * BN;

  v8f acc[2][2] = {};

  auto stage = [&](int kb0, int buf) {
    {  // A tile: BM x BK, 4 x b128 per thread
      const float* gA = A + (size_t)bm0 * lda + kb0;
#pragma unroll
      for (int i = 0; i < 4; ++i) {
        const int idx = tid + i * 256;            // 0..1023
        const int m = idx >> 3, k4 = (idx & 7) << 2;
        async_cp16(lds_off_u32(&smem[buf * AS_STRIDE + m * BKP + k4]),
                   gA + (size_t)m * lda + k4);
      }
    }
    if (BT) {  // B tile A-style: BN rows x BK, 2 x b128 per thread
      const float* gB = B + (size_t)bn0 * ldb + kb0;
#pragma unroll
      for (int i = 0; i < 2; ++i) {
        const int idx = tid + i * 256;            // 0..511
        const int n = idx >> 3, k4 = (idx & 7) << 2;
        async_cp16(lds_off_u32(&smem[BS_OFF + buf * BS_STRIDE + n * BKP + k4]),
                   gB + (size_t)n * ldb + k4);
      }
    } else {   // B tile: BK x BN, 2 x b128 per thread
      const float* gB = B + (size_t)kb0 * ldb + bn0;
#pragma unroll
      for (int i = 0; i < 2; ++i) {
        const int idx = tid + i * 256;            // 0..511
        const int kk = idx >> 4, n4 = (idx & 15) << 2;
        async_cp16(lds_off_u32(&smem[BS_OFF + buf * BS_STRIDE + kk * BN + n4]),
                   gB + (size_t)kk * ldb + n4);
      }
    }
  };

  stage(0, 0);
  async_wait0();
  __syncthreads();

  const int KB = K / BK;
  for (int kb = 0; kb < KB; ++kb) {
    if (kb + 1 < KB) stage((kb + 1) * BK, (kb + 1) & 1);  // overlap copy
    const float* As = &smem[(kb & 1) * AS_STRIDE];
    const float* Bs = &smem[BS_OFF + (kb & 1) * BS_STRIDE];
#pragma unroll
    for (int ks = 0; ks < BK; ks += 4) {
      const int ka = ks + (lh << 1);
      v2f a[2], b[2];
#pragma unroll
      for (int t = 0; t < 2; ++t)   // ds_load_b64, conflict-free (stride 36)
        a[t] = *(const v2f*)&As[(wr * 32 + t * 16 + lr) * BKP + ka];
#pragma unroll
      for (int t = 0; t < 2; ++t) {
        if (BT) {
          b[t] = *(const v2f*)&Bs[(wc * 32 + t * 16 + lr) * BKP + ka];
        } else {
          b[t].x = Bs[ka * BN + wc * 32 + t * 16 + lr];
          b[t].y = Bs[(ka + 1) * BN + wc * 32 + t * 16 + lr];
        }
      }
#pragma unroll
      for (int tm = 0; tm < 2; ++tm)
#pragma unroll
        for (int tn = 0; tn < 2; ++tn)
          acc[tm][tn] = __builtin_amdgcn_wmma_f32_16x16x4_f32(
              false, a[tm], false, b[tn], (short)0, acc[tm][tn], false, false);
    }
    async_wait0();     // next buffer resident
    __syncthreads();   // everyone done reading current buffer
  }

  // C/D layout: VGPR r, lane l -> row r + 8*(l/16), col l%16
#pragma unroll
  for (int tm = 0; tm < 2; ++tm) {
#pragma unroll
    for (int tn = 0; tn < 2; ++tn) {
      const int nn = bn0 + wc * 32 + tn * 16 + lr;
#pragma unroll
      for (int r = 0; r < 8; ++r) {
        const int m = bm0 + wr * 32 + tm * 16 + r + (lh << 3);
        float v = acc[tm][tn][r];
        if (EPI == 1) {
          v += B[(size_t)m * ldb + nn];  // + identity (A_hat = M + I part)
          v = v * dvec[m] + bias[nn];
          v = v > 0.f ? v : 0.f;
        }
        C[(size_t)m * ldc + nn] = v;
      }
    }
  }
}

// --------------------------- small support kernels -------------------------

__global__ void init_kernel(float* out, unsigned* maxD) {
  const int i = blockIdx.x * blockDim.x + threadIdx.x;
  if (i < 2 * HDIM) out[i] = 0.f;
  if (i == 0) *maxD = 0u;
}

// zero-pad feature 4096x500 -> 4096x512
__global__ void pad_feat_kernel(const float* __restrict__ f, float* __restrict__ fp) {
  const int idx = blockIdx.x * blockDim.x + threadIdx.x;
  if (idx >= NGCN * FDIMP) return;
  const int r = idx >> 9, c = idx & (FDIMP - 1);
  fp[idx] = (c < FDIM) ? f[r * FDIM + c] : 0.f;
}
// zero-pad W1 500x256 -> 512x256
__global__ void pad_w1_kernel(const float* __restrict__ w, float* __restrict__ wp) {
  const int idx = blockIdx.x * blockDim.x + threadIdx.x;
  if (idx >= FDIMP * HDIM) return;
  const int r = idx >> 8, c = idx & (HDIM - 1);
  wp[idx] = (r < FDIM) ? w[r * HDIM + c] : 0.f;
}

// sq[i] = ||x_i||^2 over padded features (zeros contribute nothing)
__global__ void sqnorm_kernel(const float* __restrict__ x, float* __restrict__ sq,
                              int F) {
  __shared__ float red[8];
  const int row = blockIdx.x;
  const float* __restrict__ r = x + (size_t)row * F;
  float s = 0.f;
  for (int j = threadIdx.x; j < F; j += blockDim.x) { float v = r[j]; s += v * v; }
  for (int o = 16; o; o >>= 1) s += __shfl_xor(s, o, 32);
  if ((threadIdx.x & 31) == 0) red[threadIdx.x >> 5] = s;
  __syncthreads();
  if (threadIdx.x == 0) {
    float t = 0.f;
    for (int w = 0; w < (int)(blockDim.x >> 5); ++w) t += red[w];
    sq[row] = t;
  }
}

// global max of D[i,j] = sq_i + sq_j - 2*XXt[i,j]  (uint atomic-max, D >= 0)
__global__ void maxd_kernel(const float* __restrict__ XXt,
                            const float* __restrict__ sq,
                            unsigned* __restrict__ maxD, int n) {
  __shared__ float red[8];
  const size_t total = (size_t)n * n;
  float m = 0.f;
  for (size_t idx = (size_t)blockIdx.x * blockDim.x + threadIdx.x; idx < total;
       idx += (size_t)gridDim.x * blockDim.x) {
    const int i = (int)(idx / n), j = (int)(idx % n);
    m = fmaxf(m, sq[i] + sq[j] - 2.f * XXt[idx]);
  }
  for (int o = 16; o; o >>= 1) m = fmaxf(m, __shfl_xor(m, o, 32));
  if ((threadIdx.x & 31) == 0) red[threadIdx.x >> 5] = m;
  __syncthreads();
  if (threadIdx.x == 0) {
    for (int w = 1; w < (int)(blockDim.x >> 5); ++w) m = fmaxf(m, red[w]);
    atomicMax(maxD, __float_as_uint(fmaxf(m, 0.f)));
  }
}

// M[i,j] = 1 iff i > j and D[i,j] < 0.5*maxD   (overwrites XXt in place)
__global__ void build_adj_kernel(float* __restrict__ XM,
                                 const float* __restrict__ sq,
                                 const unsigned* __restrict__ maxD, int n) {
  const float thr = 0.5f * __uint_as_float(*maxD);
  const size_t total = (size_t)n * n;
  for (size_t idx = (size_t)blockIdx.x * blockDim.x + threadIdx.x; idx < total;
       idx += (size_t)gridDim.x * blockDim.x) {
    const int i = (int)(idx / n), j = (int)(idx % n);
    const float D = sq[i] + sq[j] - 2.f * XM[idx];
    XM[idx] = (i > j && D < thr) ? 1.f : 0.f;
  }
}

// dv[i] = rsqrt(1 + rowsum(M_i))
__global__ void rowdeg_kernel(const float* __restrict__ M, float* __restrict__ dv,
                              int n) {
  __shared__ float red[8];
  const int row = blockIdx.x;
  const float* __restrict__ r = M + (size_t)row * n;
  float s = 0.f;
  for (int j = threadIdx.x; j < n; j += blockDim.x) s += r[j];
  for (int o = 16; o; o >>= 1) s += __shfl_xor(s, o, 32);
  if ((threadIdx.x & 31) == 0) red[threadIdx.x >> 5] = s;
  __syncthreads();
  if (threadIdx.x == 0) {
    for (int w = 1; w < (int)(blockDim.x >> 5); ++w) s += red[w];
    dv[row] = rsqrtf(s + 1.0f);
  }
}

__global__ void rowscale_kernel(float* __restrict__ y, const float* __restrict__ dv,
                                int n) {
  const int idx = blockIdx.x * blockDim.x + threadIdx.x;
  if (idx < n * HDIM) y[idx] *= dv[idx / HDIM];
}

// score[i] = tanh(dot(agg_i, Wr) + dot(x_i, Ws) + br)  (one wave per row)
__global__ void score_kernel(const float* __restrict__ agg,
                             const float* __restrict__ x,
                             const float* __restrict__ Wr,
                             const float* __restrict__ Ws,
                             const float* __restrict__ br,
                             float* __restrict__ score, int n) {
  const int row = blockIdx.x * (blockDim.x >> 5) + (threadIdx.x >> 5);
  const int lane = threadIdx.x & 31;
  if (row >= n) return;
  float s = 0.f;
  for (int h = lane; h < HDIM; h += 32)
    s += agg[(size_t)row * HDIM + h] * Wr[h] + x[(size_t)row * HDIM + h] * Ws[h];
  for (int o = 16; o; o >>= 1) s += __shfl_xor(s, o, 32);
  if (lane == 0) score[row] = tanhf(s + br[0]);
}

// exact top-k via strict rank counting (set equals reference; final result is
// order-invariant: readout is max/mean, GCN is permutation-equivariant)
__global__ void topk_kernel(const float* __restrict__ score, int* __restrict__ perm,
                            int n, int k) {
  const int i = blockIdx.x * blockDim.x + threadIdx.x;
  if (i >= n) return;
  const float si = score[i];
  int cnt = 0;
  for (int j = 0; j < n; ++j) {
    const float sj = score[j];
    cnt += (sj > si) || (sj == si && j < i);
  }
  if (cnt < k) perm[cnt] = i;
}

__global__ void gather_x_kernel(const float* __restrict__ x,
                                const float* __restrict__ score,
                                const int* __restrict__ perm,
                                float* __restrict__ xs, int k) {
  const int idx = blockIdx.x * blockDim.x + threadIdx.x;
  if (idx >= k * HDIM) return;
  const int p = idx / HDIM, h = idx % HDIM;
  const int src = perm[p];
  xs[idx] = x[(size_t)src * HDIM + h] * score[src];
}

__global__ void gather_m_kernel(const float* __restrict__ M,
                                const int* __restrict__ perm,
                                float* __restrict__ Ms, int n, int k) {
  const size_t total = (size_t)k * k;
  for (size_t idx = (size_t)blockIdx.x * blockDim.x + threadIdx.x; idx < total;
       idx += (size_t)gridDim.x * blockDim.x) {
    const int p = (int)(idx / k), q = (int)(idx % k);
    Ms[idx] = M[(size_t)perm[p] * n + perm[q]];
  }
}

// out[h] += max_r xs[r,h];  out[H+h] += mean_r xs[r,h]
__global__ void readout_kernel(const float* __restrict__ xs, float* __restrict__ out,
                               int krows) {
  __shared__ float smx[8], ssm[8];
  const int h = blockIdx.x;
  float mx = -3.402823466e38f, sm = 0.f;
  for (int r = threadIdx.x; r < krows; r += blockDim.x) {
    const float v = xs[(size_t)r * HDIM + h];
    mx = fmaxf(mx, v);
    sm += v;
  }
  for (int o = 16; o; o >>= 1) {
    mx = fmaxf(mx, __shfl_xor(mx, o, 32));
    sm += __shfl_xor(sm, o, 32);
  }
  if ((threadIdx.x & 31) == 0) { smx[threadIdx.x >> 5] = mx; ssm[threadIdx.x >> 5] = sm; }
  __syncthreads();
  if (threadIdx.x == 0) {
    for (int w = 1; w < (int)(blockDim.x >> 5); ++w) {
      mx = fmaxf(mx, smx[w]);
      sm += ssm[w];
    }
    out[h] += mx;
    out[HDIM + h] += sm / (float)krows;
  }
}

// ---------------------------------------------------------------------------
extern "C" void kernel_launch(void* const* d_in, const int* in_sizes, int n_in,
                              void* d_out, int out_size, void* d_ws, size_t ws_size,
                              hipStream_t stream) {
  (void)in_sizes; (void)n_in; (void)out_size;

  const float* feat = (const float*)d_in[0];
  const float* W[3]  = {(const float*)d_in[1], (const float*)d_in[3], (const float*)d_in[5]};
  const float* bb[3] = {(const float*)d_in[2], (const float*)d_in[4], (const float*)d_in[6]};
  const float* Wr[3] = {(const float*)d_in[7], (const float*)d_in[10], (const float*)d_in[13]};
  const float* br[3] = {(const float*)d_in[8], (const float*)d_in[11], (const float*)d_in[14]};
  const float* Ws[3] = {(const float*)d_in[9], (const float*)d_in[12], (const float*)d_in[15]};
  float* out = (float*)d_out;

  // ---- workspace layout (~127 MB) ----
  char* ws = (char*)d_ws;
  size_t off = 0;
  float* M0 = (float*)(ws + off); off += (size_t)NGCN * NGCN * 4;     // 64 MiB
  float* M1 = (float*)(ws + off); off += (size_t)3072 * 3072 * 4;     // 36 MiB
  float* xa = (float*)(ws + off); off += (size_t)NGCN * HDIM * 4;     // pooled x
  float* xb = (float*)(ws + off); off += (size_t)NGCN * HDIM * 4;     // xW / y / agg
  float* xc = (float*)(ws + off); off += (size_t)NGCN * HDIM * 4;     // conv out
  float* featP = (float*)(ws + off); off += (size_t)NGCN * FDIMP * 4; // padded feat
  float* W1P = (float*)(ws + off); off += (size_t)FDIMP * HDIM * 4;   // padded W1
  float* sq = (float*)(ws + off); off += (size_t)NGCN * 4;
  float* dv = (float*)(ws + off); off += (size_t)NGCN * 4;
  float* sc = (float*)(ws + off); off += (size_t)NGCN * 4;
  int*   pm = (int*)  (ws + off); off += (size_t)NGCN * 4;
  unsigned* mx = (unsigned*)(ws + off); off += 256;
  if (ws_size < off) return;  // insufficient scratch: refuse to scribble

  const dim3 blk(256);
  auto gemm = [&](int BT, int EPI, const float* A, int lda, const float* B, int ldb,
                  float* C, int ldc, int Mr, int Nc, int K,
                  const float* dvp, const float* bp) {
    const dim3 grd(Nc / BN, Mr / BM);
    if (BT)
      wmma_gemm_f32<true, 0><<<grd, blk, 0, stream>>>(A, lda, B, ldb, C, ldc, K, dvp, bp);
    else if (EPI == 0)
      wmma_gemm_f32<false, 0><<<grd, blk, 0, stream>>>(A, lda, B, ldb, C, ldc, K, dvp, bp);
    else
      wmma_gemm_f32<false, 1><<<grd, blk, 0, stream>>>(A, lda, B, ldb, C, ldc, K, dvp, bp);
  };

  // ---- adjacency build ----
  init_kernel<<<2, blk, 0, stream>>>(out, mx);
  pad_feat_kernel<<<(NGCN * FDIMP + 255) / 256, blk, 0, stream>>>(feat, featP);
  pad_w1_kernel<<<(FDIMP * HDIM + 255) / 256, blk, 0, stream>>>(W[0], W1P);
  sqnorm_kernel<<<NGCN, blk, 0, stream>>>(featP, sq, FDIMP);
  // XXt = feat @ feat^T  (16.8 GFLOP fp32 WMMA)
  gemm(1, 0, featP, FDIMP, featP, FDIMP, M0, NGCN, NGCN, NGCN, FDIMP, nullptr, nullptr);
  maxd_kernel<<<1024, blk, 0, stream>>>(M0, sq, mx, NGCN);
  build_adj_kernel<<<2048, blk, 0, stream>>>(M0, sq, mx, NGCN);

  // ---- three GCN + SAGPool layers ----
  const int nArr[4] = {4096, 3072, 2304, 1728};  // ceil(0.75*n) chain
  const float* Mcur = M0;
  float* Mnext = M1;
  for (int l = 0; l < 3; ++l) {
    const int n = nArr[l], k = nArr[l + 1];
    const float* xin = (l == 0) ? featP : xa;
    const float* Wl  = (l == 0) ? W1P : W[l];
    const int Kin = (l == 0) ? FDIMP : HDIM;

    rowdeg_kernel<<<n, blk, 0, stream>>>(Mcur, dv, n);
    // z = x @ W
    gemm(0, 0, xin, Kin, Wl, HDIM, xb, HDIM, n, HDIM, Kin, nullptr, nullptr);
    // y = d .* z  (in place)
    rowscale_kernel<<<(n * HDIM + 255) / 256, blk, 0, stream>>>(xb, dv, n);
    // xc = relu(d .* (M@y + y) + b)   (fused GCN epilogue, K = n)
    gemm(0, 1, Mcur, n, xb, HDIM, xc, HDIM, n, HDIM, n, dv, bb[l]);
    // agg = M @ xc  (pooling attention)
    gemm(0, 0, Mcur, n, xc, HDIM, xb, HDIM, n, HDIM, n, nullptr, nullptr);
    score_kernel<<<(n + 7) / 8, blk, 0, stream>>>(xb, xc, Wr[l], Ws[l], br[l], sc, n);
    topk_kernel<<<(n + 255) / 256, blk, 0, stream>>>(sc, pm, n, k);
    gather_x_kernel<<<(k * HDIM + 255) / 256, blk, 0, stream>>>(xc, sc, pm, xa, k);
    if (l < 2)
      gather_m_kernel<<<2048, blk, 0, stream>>>(Mcur, pm, Mnext, n, k);
    readout_kernel<<<HDIM, blk, 0, stream>>>(xa, out, k);

    const float* t = Mcur;
    Mcur = Mnext;
    Mnext = (float*)t;
  }
}